// BasicBlock_37520834297872
// MI455X (gfx1250) — compile-verified
//
#include <hip/hip_runtime.h>
#include <hip/hip_bf16.h>
#include <math.h>

// Problem constants (match reference)
#define A_ANG  64
#define DC     1024
#define H_IMG  256
#define W_IMG  256
#define NSAMP  512
#define NPAD   2048
#define FEAT   32
#define DSRC   500.0f
#define DDET   500.0f
#define SPACING 2.0f
#define PIXELS (H_IMG*W_IMG)

typedef float v8f  __attribute__((ext_vector_type(8)));
typedef float v2f  __attribute__((ext_vector_type(2)));
typedef unsigned int u32x4 __attribute__((ext_vector_type(4)));
typedef int   i32x8 __attribute__((ext_vector_type(8)));
typedef int   i32x4 __attribute__((ext_vector_type(4)));

// ---------------------------------------------------------------------------
// CDNA5 Tensor Data Mover: 2D tile (dim0 x dim1 of 4-byte elems) global->LDS.
// Descriptor per CDNA5 ISA (group0: count/lds/global/type,
// group1: data_size + dims/strides). Groups 2/3 zero (2D tensor). This
// toolchain exposes the 6-arg builtin: (g0, g1, g2, g3, g4, cpol).
// ---------------------------------------------------------------------------
__device__ __forceinline__ void tdm_load_2d(unsigned lds_byte_off,
                                            const void* gptr,
                                            unsigned dim0, unsigned dim1,
                                            unsigned stride0_elems) {
  unsigned long long ga = (unsigned long long)gptr;
  u32x4 g0;
  g0.x = 1u;                                      // count=1, user descriptor
  g0.y = lds_byte_off;                            // lds_addr (bytes)
  g0.z = (unsigned)(ga & 0xFFFFFFFFu);            // global_addr[31:0]
  g0.w = (unsigned)((ga >> 32) & 0x1FFFFFFu)      // global_addr[56:32]
       | (2u << 30);                              // type = 2 ("image")
  i32x8 g1;
  g1[0] = (int)(2u << 16);                        // data_size=2 (4B), mask=0
  g1[1] = (int)((dim0 & 0xFFFFu) << 16);          // tensor_dim0[15:0]
  g1[2] = (int)((dim0 >> 16) | ((dim1 & 0xFFFFu) << 16));
  g1[3] = (int)((dim1 >> 16) | ((dim0 & 0xFFFFu) << 16)); // tile_dim0 = dim0
  g1[4] = (int)(dim1 & 0xFFFFu);                  // tile_dim1 = dim1, tile_dim2=0
  g1[5] = (int)stride0_elems;                     // tensor_dim0_stride[31:0]
  g1[6] = 0;
  g1[7] = 0;
  i32x4 z4 = {0, 0, 0, 0};
  i32x8 z8 = {0, 0, 0, 0, 0, 0, 0, 0};
  __builtin_amdgcn_tensor_load_to_lds(g0, g1, z4, z4, z8, 0);
}

// ---------------------------------------------------------------------------
// Ramp filter impulse response: h = irfft(2*|rfftfreq(2048)|).
// h[m] = (4/N^2) * sum_{j=1}^{N/2-1} j*cos(2*pi*j*m/N) + (1/N)*cos(pi*m)
// ---------------------------------------------------------------------------
__global__ void ramp_kernel(float* __restrict__ h) {
  int m = blockIdx.x * blockDim.x + threadIdx.x;
  if (m >= NPAD) return;
  const float w = 2.0f * 3.14159265358979323846f / (float)NPAD;
  float acc = 0.0f;
  for (int j = 1; j < NPAD / 2; ++j) {
    int ph = (j * m) & (NPAD - 1);                // exact phase reduction
    acc += (float)j * __cosf(w * (float)ph);
  }
  float val = (4.0f / ((float)NPAD * (float)NPAD)) * acc
            + (1.0f / (float)NPAD) * ((m & 1) ? -1.0f : 1.0f);
  h[m] = val;
}

// ---------------------------------------------------------------------------
// Fan-beam forward projection. Whole image staged into LDS (256KB < 320KB/WGP)
// via TDM; one thread per ray, 512 bilinear samples from LDS.
// ---------------------------------------------------------------------------
__global__ void fwdproj_kernel(const float* __restrict__ img,
                               const float* __restrict__ theta,
                               float* __restrict__ sino) {
  extern __shared__ float lim[];                  // H_IMG*W_IMG floats
  if ((threadIdx.x >> 5) == 0) {                  // wave 0 issues the TDM op
    tdm_load_2d(0u, img, W_IMG, H_IMG, W_IMG);
    __builtin_amdgcn_s_wait_tensorcnt(0);
  }
  __syncthreads();

  int r  = blockIdx.x * blockDim.x + threadIdx.x; // ray id
  int a  = r >> 10;
  int dc = r & (DC - 1);
  float c, s;
  __sincosf(theta[a], &s, &c);
  float u  = ((float)dc - (DC - 1) * 0.5f) * SPACING;
  float sx = DSRC * c, sy = DSRC * s;
  float dx = -(DSRC + DDET) * c - u * s;          // det - src
  float dy = -(DSRC + DDET) * s + u * c;
  float L  = sqrtf((DSRC + DDET) * (DSRC + DDET) + u * u);
  const float inv = 1.0f / (float)NSAMP;
  float acc = 0.0f;
  for (int i = 0; i < NSAMP; ++i) {
    float t  = ((float)i + 0.5f) * inv;
    float cx = sx + t * dx + (W_IMG - 1) * 0.5f;
    float cy = sy + t * dy + (H_IMG - 1) * 0.5f;
    float x0f = floorf(cx), y0f = floorf(cy);
    float fx = cx - x0f, fy = cy - y0f;
    int x0 = (int)x0f, y0 = (int)y0f;
    float v00 = ((unsigned)x0     < W_IMG && (unsigned)y0     < H_IMG) ? lim[y0 * W_IMG + x0]           : 0.f;
    float v10 = ((unsigned)(x0+1) < W_IMG && (unsigned)y0     < H_IMG) ? lim[y0 * W_IMG + x0 + 1]       : 0.f;
    float v01 = ((unsigned)x0     < W_IMG && (unsigned)(y0+1) < H_IMG) ? lim[(y0 + 1) * W_IMG + x0]     : 0.f;
    float v11 = ((unsigned)(x0+1) < W_IMG && (unsigned)(y0+1) < H_IMG) ? lim[(y0 + 1) * W_IMG + x0 + 1] : 0.f;
    acc += (v00 * (1.f - fx) + v10 * fx) * (1.f - fy)
         + (v01 * (1.f - fx) + v11 * fx) * fy;
  }
  sino[r] = acc * L * inv;
}

// ---------------------------------------------------------------------------
// Ramp filtering as GEMM on the f32 WMMA pipe:
//   filtered[a,k] = sum_n sino[a,n] * h[(k-n) mod 2048], then - measured sino.
// M=64 angles, N=1024 detectors, K=1024. v_wmma_f32_16x16x4_f32 per 4 K-steps.
// A layout (16x4 f32): lanes 0-15 rows, halves select K pairs (ISA 7.12.2).
// ---------------------------------------------------------------------------
__global__ void filter_gemm_kernel(const float* __restrict__ sino,
                                   const float* __restrict__ hflt,
                                   const float* __restrict__ meas,
                                   float* __restrict__ diff) {
  __shared__ float sh[NPAD];
  for (int i = threadIdx.x; i < NPAD; i += blockDim.x) sh[i] = hflt[i];
  __syncthreads();

  int wv   = (blockIdx.x * blockDim.x + threadIdx.x) >> 5;
  int lane = threadIdx.x & 31;
  int mt   = wv >> 6;                             // angle tile 0..3
  int nt   = wv & 63;                             // detector tile 0..63
  int half = lane >> 4;
  int arow = mt * 16 + (lane & 15);               // A row (angle)
  int n    = nt * 16 + (lane & 15);               // B/D column (detector)
  v8f acc = {0, 0, 0, 0, 0, 0, 0, 0};
  for (int kt = 0; kt < DC; kt += 4) {
    int k0 = kt + 2 * half;
    v2f av, bv;
    av.x = sino[arow * DC + k0];
    av.y = sino[arow * DC + k0 + 1];
    bv.x = sh[(n - k0)     & (NPAD - 1)];
    bv.y = sh[(n - k0 - 1) & (NPAD - 1)];
    acc = __builtin_amdgcn_wmma_f32_16x16x4_f32(false, av, false, bv,
                                                (short)0, acc, false, false);
  }
  for (int v = 0; v < 8; ++v) {
    int row = mt * 16 + v + 8 * half;             // D layout: M = v + 8*half
    diff[row * DC + n] = acc[v] - meas[row * DC + n];
  }
}

// ---------------------------------------------------------------------------
// Pixel-driven fan-beam backprojection. Residual sinogram (64x1024 f32 =
// 256KB) staged into LDS via TDM. Fused: x_input = x - lambda * bp.
// ---------------------------------------------------------------------------
__global__ void backproj_kernel(const float* __restrict__ diff,
                                const float* __restrict__ theta,
                                const float* __restrict__ x_in,
                                const float* __restrict__ lambda_p,
                                float* __restrict__ x_input) {
  extern __shared__ float lsino[];                // A_ANG*DC floats
  if ((threadIdx.x >> 5) == 0) {
    tdm_load_2d(0u, diff, DC, A_ANG, DC);
    __builtin_amdgcn_s_wait_tensorcnt(0);
  }
  __syncthreads();

  int pix = blockIdx.x * blockDim.x + threadIdx.x;
  int y = pix >> 8, x = pix & 255;
  float gx = (float)x - (W_IMG - 1) * 0.5f;
  float gy = (float)y - (H_IMG - 1) * 0.5f;
  float acc = 0.0f;
  for (int a = 0; a < A_ANG; ++a) {
    float c, s;
    __sincosf(theta[a], &s, &c);
    float xr = gx * c + gy * s;
    float yr = -gx * s + gy * c;
    float invd = 1.0f / (DSRC - xr);
    float uu  = yr * (DSRC + DDET) * invd;
    float wgt = (DSRC * DSRC) * invd * invd;
    float iu  = uu / SPACING + (DC - 1) * 0.5f;
    float i0f = floorf(iu);
    float fr  = iu - i0f;
    int   i0  = (int)i0f;
    float v0 = ((unsigned)i0       < DC) ? lsino[a * DC + i0]     : 0.f;
    float v1 = ((unsigned)(i0 + 1) < DC) ? lsino[a * DC + i0 + 1] : 0.f;
    acc += (v0 * (1.f - fr) + v1 * fr) * wgt;
  }
  float bp = acc * (3.14159265358979323846f / (float)A_ANG);
  x_input[pix] = x_in[pix] - lambda_p[0] * bp;
}

// ---------------------------------------------------------------------------
// 3x3 SAME conv as implicit GEMM on f32 WMMA:
//   M = 65536 pixels (16-pixel row tiles), N = Cout, K = Cin*9 (padded to 4).
// Weights (K x 32, zero-padded) staged in LDS. Fused epilogues.
// ---------------------------------------------------------------------------
#define CMODE_NONE 0
#define CMODE_RELU 1
#define CMODE_SOFT 2   // writes raw to out, soft-threshold to out2
#define CMODE_SUB  3   // writes out = val - sub (symloss)

__device__ __forceinline__ float fetchA_conv(const float* __restrict__ in,
                                             int K, int k, int y, int xs) {
  if (k >= K) return 0.0f;
  int ci = k / 9;  int r9 = k - ci * 9;
  int dy = r9 / 3; int dxx = r9 - dy * 3;
  int sy = y + dy - 1, sx = xs + dxx - 1;
  if ((unsigned)sy < (unsigned)H_IMG && (unsigned)sx < (unsigned)W_IMG)
    return in[ci * PIXELS + sy * W_IMG + sx];
  return 0.0f;
}

__global__ void conv_wmma_kernel(const float* __restrict__ in, int Cin,
                                 const float* __restrict__ w, int Cout,
                                 float* __restrict__ out,
                                 float* __restrict__ out2,
                                 const float* __restrict__ sub,
                                 const float* __restrict__ thr_p,
                                 int mode, int Ntiles) {
  __shared__ float wl[FEAT * 9 * FEAT];           // up to 288x32 f32 = 36KB
  int K  = Cin * 9;
  int Kp = (K + 3) & ~3;
  for (int i = threadIdx.x; i < Kp * 32; i += blockDim.x) {
    int k = i >> 5, co = i & 31;
    wl[i] = (k < K && co < Cout) ? w[co * K + k] : 0.0f;
  }
  __syncthreads();

  int wv   = blockIdx.x * (blockDim.x >> 5) + (threadIdx.x >> 5);
  int lane = threadIdx.x & 31;
  int mt   = wv / Ntiles;
  int nt   = wv - mt * Ntiles;
  int p0   = mt * 16;                             // 16 consecutive pixels (row-aligned)
  int y    = p0 >> 8;
  int xs   = (p0 & 255) + (lane & 15);            // A-row pixel x
  int half = lane >> 4;
  int ncol = nt * 16 + (lane & 15);               // output channel column
  v8f acc = {0, 0, 0, 0, 0, 0, 0, 0};
  for (int kt = 0; kt < Kp; kt += 4) {
    int k0 = kt + 2 * half;
    v2f av, bv;
    av.x = fetchA_conv(in, K, k0,     y, xs);
    av.y = fetchA_conv(in, K, k0 + 1, y, xs);
    bv.x = wl[k0 * 32 + ncol];
    bv.y = wl[(k0 + 1) * 32 + ncol];
    acc = __builtin_amdgcn_wmma_f32_16x16x4_f32(false, av, false, bv,
                                                (short)0, acc, false, false);
  }
  if (ncol < Cout) {
    float thr = (mode == CMODE_SOFT) ? thr_p[0] : 0.0f;
    for (int v = 0; v < 8; ++v) {
      int pix  = p0 + v + 8 * half;               // D layout: M = v + 8*half
      int oidx = ncol * PIXELS + pix;
      float val = acc[v];
      if (mode == CMODE_RELU) {
        out[oidx] = fmaxf(val, 0.0f);
      } else if (mode == CMODE_SOFT) {
        out[oidx]  = val;
        out2[oidx] = copysignf(fmaxf(fabsf(val) - thr, 0.0f), val);
      } else if (mode == CMODE_SUB) {
        out[oidx] = val - sub[pix];
      } else {
        out[oidx] = val;
      }
    }
  }
}

// ---------------------------------------------------------------------------
extern "C" void kernel_launch(void* const* d_in, const int* in_sizes, int n_in,
                              void* d_out, int out_size, void* d_ws, size_t ws_size,
                              hipStream_t stream) {
  const float* x        = (const float*)d_in[0];  // (1,1,256,256)
  const float* theta    = (const float*)d_in[1];  // (64)
  const float* sino_ms  = (const float*)d_in[2];  // (1,64,1024)
  const float* lambda_p = (const float*)d_in[3];  // scalar
  const float* thr_p    = (const float*)d_in[4];  // scalar
  const float* w1f      = (const float*)d_in[5];  // (32,1,3,3)
  const float* w2f      = (const float*)d_in[6];  // (32,32,3,3)
  const float* w1b      = (const float*)d_in[7];  // (32,32,3,3)
  const float* w2b      = (const float*)d_in[8];  // (1,32,3,3)
  float* out = (float*)d_out;                     // [x_pred | symloss]
  float* ws  = (float*)d_ws;

  float* h         = ws;                          // 2048
  float* sino_pred = ws + 4096;                   // 65536
  float* diff      = sino_pred + A_ANG * DC;      // 65536
  float* x_input   = diff + A_ANG * DC;           // 65536
  float* feat1     = x_input + PIXELS;            // 32*65536
  float* x_fwd     = feat1 + FEAT * PIXELS;       // 32*65536
  float* hs        = x_fwd + FEAT * PIXELS;       // 32*65536
  float* t         = hs + FEAT * PIXELS;          // 32*65536

  // 1) ramp filter table
  ramp_kernel<<<NPAD / 256, 256, 0, stream>>>(h);
  // 2) forward projection (image in LDS via TDM)
  fwdproj_kernel<<<(A_ANG * DC) / 256, 256, PIXELS * sizeof(float), stream>>>(
      x, theta, sino_pred);
  // 3) ramp filtering as WMMA GEMM, fused residual (filtered - sinogram)
  filter_gemm_kernel<<<32, 256, 0, stream>>>(sino_pred, h, sino_ms, diff);
  // 4) backprojection (residual sinogram in LDS via TDM), fused x_input
  backproj_kernel<<<PIXELS / 1024, 1024, A_ANG * DC * sizeof(float), stream>>>(
      diff, theta, x, lambda_p, x_input);
  // 5) conv chain (implicit GEMM on f32 WMMA)
  conv_wmma_kernel<<<1024, 256, 0, stream>>>(x_input, 1,  w1f, 32, feat1, nullptr, nullptr, thr_p, CMODE_RELU, 2);
  conv_wmma_kernel<<<1024, 256, 0, stream>>>(feat1,  32, w2f, 32, x_fwd, hs,     nullptr, thr_p, CMODE_SOFT, 2);
  conv_wmma_kernel<<<1024, 256, 0, stream>>>(hs,     32, w1b, 32, t,     nullptr, nullptr, thr_p, CMODE_RELU, 2);
  conv_wmma_kernel<<< 512, 256, 0, stream>>>(t,      32, w2b, 1,  out,   nullptr, nullptr, thr_p, CMODE_NONE, 1);
  conv_wmma_kernel<<<1024, 256, 0, stream>>>(x_fwd,  32, w1b, 32, t,     nullptr, nullptr, thr_p, CMODE_RELU, 2);
  conv_wmma_kernel<<< 512, 256, 0, stream>>>(t,      32, w2b, 1,  out + PIXELS, nullptr, x_input, thr_p, CMODE_SUB, 1);
}